// BodyFst_506806141153
// MI455X (gfx1250) — compile-verified
//
#include <hip/hip_runtime.h>
#include <hip/hip_bf16.h>

typedef __attribute__((ext_vector_type(2))) float v2f;
typedef __attribute__((ext_vector_type(8))) float v8f;

// Raw 32-bit LDS byte offset for an extern-__shared__ element:
// generic -> addrspace(3) addrspacecast (strips aperture), then ptrtoint.
typedef __attribute__((address_space(3))) const void* lds_cptr_t;
static __device__ inline unsigned lds_off(const void* p) {
    return (unsigned)(uintptr_t)(lds_cptr_t)p;
}

// One workgroup handles one (n, h) slice:
//   Phase A: async-DMA L[:,h,:] and R[:,h,:] (C x W each) straight into LDS
//            (GLOBAL_LOAD_ASYNC_TO_LDS_B32, ASYNCcnt), transposed with pad.
//   Phase B: per-x inverse L2 norms into LDS.
//   Phase C: 8 waves compute 16x16 gram tiles with V_WMMA_F32_16X16X4_F32,
//            fold norms in afterwards, transpose via per-wave LDS scratch,
//            and store coalesced diagonal rows of the cost volume.
__global__ __launch_bounds__(256) void cost_volume_wmma(
    const float* __restrict__ L, const float* __restrict__ R,
    float* __restrict__ out, int C, int H, int W, int D)
{
    extern __shared__ float smem[];
    const int CP = C + 2;                 // padded channel stride (even -> b64 aligned)
    float* sL   = smem;                   // CP*W  : sL[x*CP + c]
    float* sR   = sL + CP * W;            // CP*W
    float* invL = sR + CP * W;            // W
    float* invR = invL + W;               // W
    float* sScr = invR + W;               // 8 waves * 16*17 transpose scratch

    const int tid  = threadIdx.x;
    const int lane = tid & 31;
    const int wid  = tid >> 5;
    const int n = blockIdx.x / H;
    const int h = blockIdx.x - n * H;

    const size_t cstr = (size_t)H * W;                 // channel stride in elements
    const float* Lb = L + ((size_t)n * C * H + h) * W; // element (c,x) at Lb[c*cstr + x]
    const float* Rb = R + ((size_t)n * C * H + h) * W;

    // ---- Phase A: async global->LDS staging with transpose scatter ----
    // Per lane: global addr = row base (SGPR pair) + x*4 (VGPR offset),
    //           LDS dest    = (x*CP + c)*4 (VGPR), no VGPR data round trip.
    for (int c = 0; c < C; ++c) {
        const float* lp = Lb + (size_t)c * cstr;
        const float* rp = Rb + (size_t)c * cstr;
        for (int x = tid; x < W; x += 256) {
            unsigned dl  = lds_off(&sL[x * CP + c]);
            unsigned dr  = lds_off(&sR[x * CP + c]);
            unsigned off = (unsigned)x * 4u;
            asm volatile("global_load_async_to_lds_b32 %0, %1, %2"
                         :: "v"(dl), "v"(off), "s"(lp) : "memory");
            asm volatile("global_load_async_to_lds_b32 %0, %1, %2"
                         :: "v"(dr), "v"(off), "s"(rp) : "memory");
        }
    }
    asm volatile("s_wait_asynccnt 0x0" ::: "memory");  // this wave's DMAs landed
    __syncthreads();                                   // everyone's DMAs landed

    // ---- Phase B: inverse norms (ref: v / max(||v||, 1e-12)) ----
    for (int x = tid; x < W; x += 256) {
        const float* pl = sL + x * CP;
        const float* pr = sR + x * CP;
        float a = 0.f, b = 0.f;
        for (int c = 0; c < C; ++c) { a += pl[c] * pl[c]; b += pr[c] * pr[c]; }
        invL[x] = 1.0f / fmaxf(sqrtf(a), 1e-12f);
        invR[x] = 1.0f / fmaxf(sqrtf(b), 1e-12f);
    }
    __syncthreads();

    // ---- Phase C: banded gram via WMMA ----
    float* S = sScr + wid * (16 * 17);      // per-wave 16x16 tile scratch, stride 17
    const int TX = W >> 4;                  // x tiles
    const int TJ = (D >> 4) + 1;            // diagonal tile offsets t = 0..D/16
    const int npairs = TX * TJ;
    const int m     = lane & 15;            // A row / B col within tile
    const int khalf = (lane >> 4) << 1;     // K pair base: 0 or 2 (A/B f32 layout)
    const int ab    = (lane >> 4) << 3;     // C/D row base: 0 or 8
    const int qbase = (lane >> 4) << 4;     // diagonal base for writeback halves

    for (int p = wid; p < npairs; p += 8) {
        const int xt  = p / TJ;
        const int t   = p - xt * TJ;
        const int x0  = xt << 4;
        const int xr0 = x0 - (t << 4);

        const int xA  = x0 + m;                       // left column this lane feeds
        int xB        = xr0 + m;                      // right column (may be < 0)
        const int xBc = xB < 0 ? 0 : xB;              // clamp; masked at writeback

        const float* pa = sL + xA * CP + khalf;
        const float* pb = sR + xBc * CP + khalf;

        v8f acc = {};
        for (int c = 0; c < C; c += 4) {
            v2f av = *(const v2f*)(pa + c);
            v2f bv = *(const v2f*)(pb + c);
            acc = __builtin_amdgcn_wmma_f32_16x16x4_f32(
                false, av, false, bv, (short)0, acc, false, false);
        }

        // Stage D tile into LDS: S[a][b], a = row (left), b = col (right).
        #pragma unroll
        for (int v = 0; v < 8; ++v)
            S[(v + ab) * 17 + m] = acc[v];
        // Same-wave DS ops are in-order; compiler inserts s_wait_dscnt on use.

        // Writeback by diagonals: fixed d => 16 consecutive x per half-wave.
        for (int qq = 0; qq < 16; ++qq) {
            const int q  = qq + qbase;        // diagonal id 0..31
            const int dl = q - 15;            // a - b
            const int d  = (t << 4) + dl;     // disparity
            const int a  = m;
            const int b  = a - dl;
            const int xL = x0 + a;
            if (d >= 0 && d < D && b >= 0 && b < 16) {
                float val = 0.0f;
                if (d <= xL)                  // x >= disparity: valid
                    val = S[a * 17 + b] * invL[xL] * invR[xL - d];
                out[(((size_t)n * D + d) * H + h) * W + xL] = val;
            }
        }
    }
}

extern "C" void kernel_launch(void* const* d_in, const int* in_sizes, int n_in,
                              void* d_out, int out_size, void* d_ws, size_t ws_size,
                              hipStream_t stream) {
    (void)in_sizes; (void)n_in; (void)out_size; (void)d_ws; (void)ws_size;
    const float* l0 = (const float*)d_in[0];
    const float* r0 = (const float*)d_in[1];
    const float* l1 = (const float*)d_in[2];
    const float* r1 = (const float*)d_in[3];
    const float* l2 = (const float*)d_in[4];
    const float* r2 = (const float*)d_in[5];
    float* out = (float*)d_out;

    // Allow large dynamic LDS (max need: level0 = 152064 B < 320 KB/WGP).
    hipFuncSetAttribute((const void*)cost_volume_wmma,
                        hipFuncAttributeMaxDynamicSharedMemorySize, 160 * 1024);

    auto launch = [&](const float* Lp, const float* Rp, float* O,
                      int C, int H, int W, int D) {
        const int CP = C + 2;
        const size_t shmem = (size_t)(2 * CP * W + 2 * W + 8 * 16 * 17) * sizeof(float);
        cost_volume_wmma<<<dim3(2 * H), dim3(256), shmem, stream>>>(Lp, Rp, O, C, H, W, D);
    };

    launch(l0, r0, out,            32, 256, 512, 128);
    launch(l1, r1, out + 33554432, 64, 128, 256,  64);
    launch(l2, r2, out + 37748736, 96,  64, 128,  32);
}